// RaggedGravNet_57569741635782
// MI455X (gfx1250) — compile-verified
//
#include <hip/hip_runtime.h>
#include <cfloat>
#include <cmath>

// ---------------------------------------------------------------------------
// GravNet forward for MI455X (gfx1250, wave32).
//  N=80000 nodes, 8 events x 10000, F_IN=64, D=4, P=32, K=40, out=64.
//  - kNN pairwise dots via V_WMMA_F32_16X16X4_F32, candidates-as-rows so each
//    lane keeps 8 distances of ONE query in registers: zero barriers in the
//    625-iteration candidate loop, all 32 lanes do top-K selection.
//  - dense layers via chained f32 WMMA (16x16 tiles)
//  - gather/aggregate: wave=node, lane=channel (P==32==wave32)
// ---------------------------------------------------------------------------

#define N_EVENTS 8
#define NPE      10000
#define NTOT     (N_EVENTS * NPE)
#define F_IN     64
#define NDIM     4
#define NPROP    32
#define NFILT    64
#define KNBR     40

typedef __attribute__((ext_vector_type(2))) float v2f;
typedef __attribute__((ext_vector_type(4))) float v4f;
typedef __attribute__((ext_vector_type(8))) float v8f;

__device__ __forceinline__ v8f wmma_f32_4(v2f a, v2f b, v8f c) {
  // (neg_a, A, neg_b, B, c_mod, C, reuse_a, reuse_b)
  return __builtin_amdgcn_wmma_f32_16x16x4_f32(false, a, false, b, (short)0, c,
                                               false, false);
}

// ---------------------------------------------------------------------------
// coords = x @ W_s + b_s  (N x 64)@(64 x 4), plus |coord|^2 per node.
// One thread per node; W_s (1 KB) broadcast through scalar/L0 caches.
// ---------------------------------------------------------------------------
__global__ void coords_kernel(const float* __restrict__ x,
                              const float* __restrict__ Ws,
                              const float* __restrict__ bs,
                              float* __restrict__ coords,
                              float* __restrict__ norms) {
  int t = blockIdx.x * blockDim.x + threadIdx.x;
  if (t >= NTOT) return;
  float a0 = bs[0], a1 = bs[1], a2 = bs[2], a3 = bs[3];
  const float* xr = x + (size_t)t * F_IN;
#pragma unroll 8
  for (int k = 0; k < F_IN; ++k) {
    float xv = xr[k];
    a0 += xv * Ws[k * NDIM + 0];
    a1 += xv * Ws[k * NDIM + 1];
    a2 += xv * Ws[k * NDIM + 2];
    a3 += xv * Ws[k * NDIM + 3];
  }
  float* cr = coords + (size_t)t * NDIM;
  cr[0] = a0; cr[1] = a1; cr[2] = a2; cr[3] = a3;
  norms[t] = a0 * a0 + a1 * a1 + a2 * a2 + a3 * a3;
}

// ---------------------------------------------------------------------------
// Generic WMMA GEMM: out[rb:rb+16, cb:cb+16] = act([A0|A1] @ W + bias)
//   A = horizontal concat of A0 (width w0) and A1 (width w1); K = w0+w1.
//   W row-major (K x n_out). act: 0=none, 1=elu, 2=tanh.
//   One wave per 16x16 output tile; K stepped by 4 with f32 WMMA.
// ---------------------------------------------------------------------------
__global__ void gemm_wmma_kernel(const float* __restrict__ A0, int w0,
                                 const float* __restrict__ A1, int w1,
                                 const float* __restrict__ W,
                                 const float* __restrict__ bias,
                                 float* __restrict__ out, int n_out, int act) {
  int lane = threadIdx.x & 31;
  int n = lane & 15;       // output column within tile (and A row within tile)
  int hi = lane >> 4;      // selects K pair {0,1} vs {2,3}
  int rb = blockIdx.x * 16;
  int cb = blockIdx.y * 16;
  int K = w0 + w1;

  v8f acc = {};
  for (int k0 = 0; k0 < K; k0 += 4) {
    const float* src;
    int lk, stride;
    if (k0 < w0) { src = A0; lk = k0;      stride = w0; }
    else         { src = A1; lk = k0 - w0; stride = w1; }
    // A fragment: lane n holds A[rb+n][lk + 2*hi .. +1]
    v2f a = *(const v2f*)(src + (size_t)(rb + n) * stride + lk + hi * 2);
    // B fragment: lane n holds W[k0+2*hi][cb+n], W[k0+2*hi+1][cb+n]
    v2f b;
    int col = cb + n;
    b.x = W[(size_t)(k0 + hi * 2 + 0) * n_out + col];
    b.y = W[(size_t)(k0 + hi * 2 + 1) * n_out + col];
    acc = wmma_f32_4(a, b, acc);
  }

  float bv = bias[cb + n];
#pragma unroll
  for (int r = 0; r < 8; ++r) {
    int m = r + hi * 8;  // C layout: VGPR r -> row r (lanes 0-15) / r+8 (16-31)
    float v = acc[r] + bv;
    if (act == 1)      v = (v > 0.f) ? v : (__expf(v) - 1.f);  // elu
    else if (act == 2) v = tanhf(v);
    out[(size_t)(rb + m) * n_out + cb + n] = v;
  }
}

// ---------------------------------------------------------------------------
// Brute-force per-event kNN. One wave (one block) owns a 16-query tile.
// Per chunk of 16 candidates: one WMMA gives C[cand][query]; lane L holds 8
// distances for query (L&15), candidate rows r+8*(L>>4). Each lane streams
// its half of the candidates into a PRIVATE top-40 (dist,idx) buffer in LDS
// (lane-private => in-order DS, no barriers in the hot loop). A single
// barrier + merge at the end combines the two 40-entry half-lists per query.
// Self is excluded by index compare (reference drops nearest == self).
// Neighbour order is irrelevant downstream (max/mean are permutation-inv).
// ---------------------------------------------------------------------------
__global__ void knn_kernel(const float* __restrict__ coords,
                           const float* __restrict__ norms,
                           int* __restrict__ nbr,
                           float* __restrict__ d2out) {
  __shared__ float s_bd[32][KNBR];
  __shared__ int   s_bi[32][KNBR];

  int lane = threadIdx.x & 31;
  int tile = blockIdx.x;                    // 5000 tiles (625 per event)
  int qbase  = tile * 16;
  int ev     = tile / (NPE / 16);
  int evbase = ev * NPE;
  int n  = lane & 15;
  int hi = lane >> 4;

  // Query B fragment (loop invariant): lane n holds q[qbase+n][2*hi..2*hi+1]
  v2f bq = *(const v2f*)(coords + (size_t)(qbase + n) * NDIM + hi * 2);
  float qn = norms[qbase + n];              // norm of this lane's query
  int   qg = qbase + n;                     // this lane's query global index

  // Private top-40 buffer (worst-replacement scheme)
  float worst = FLT_MAX;
  int   wpos  = 0;
#pragma unroll
  for (int t = 0; t < KNBR; ++t) {
    s_bd[lane][t] = FLT_MAX;
    s_bi[lane][t] = qg;
  }

  for (int chunk = 0; chunk < NPE / 16; ++chunk) {
    int cbase = evbase + chunk * 16;
    // Candidate A fragment: lane m holds c[cbase+m][2*hi..2*hi+1]
    v2f ac = *(const v2f*)(coords + (size_t)(cbase + n) * NDIM + hi * 2);
    // Candidate norms for this lane's 8 rows (r + 8*hi): contiguous 8 floats
    v4f cn0 = *(const v4f*)(norms + cbase + hi * 8);
    v4f cn1 = *(const v4f*)(norms + cbase + hi * 8 + 4);

    v8f dotv = {};
    dotv = wmma_f32_4(ac, bq, dotv);        // full K=4 in one WMMA

#pragma unroll
    for (int r = 0; r < 8; ++r) {
      float cn = (r < 4) ? cn0[r & 3] : cn1[r & 3];
      float d = cn + qn - 2.f * dotv[r];
      int cg = cbase + r + hi * 8;
      if (d < worst && cg != qg) {
        s_bd[lane][wpos] = d;
        s_bi[lane][wpos] = cg;
        float mx = -FLT_MAX; int mp = 0;
        for (int t = 0; t < KNBR; ++t) {
          float v = s_bd[lane][t];
          if (v > mx) { mx = v; mp = t; }
        }
        worst = mx; wpos = mp;
      }
    }
  }

  __syncthreads();  // single-wave workgroup: effectively a waitcnt, not a bar

  // Merge: lanes 0-15 fold lane+16's half-list into their own, keep top-40.
  if (lane < 16) {
    float mx = -FLT_MAX; int mp = 0;
#pragma unroll
    for (int t = 0; t < KNBR; ++t) {
      float v = s_bd[lane][t];
      if (v > mx) { mx = v; mp = t; }
    }
    worst = mx; wpos = mp;
    for (int t = 0; t < KNBR; ++t) {
      float d = s_bd[lane + 16][t];
      if (d < worst) {
        s_bd[lane][wpos] = d;
        s_bi[lane][wpos] = s_bi[lane + 16][t];
        mx = -FLT_MAX; mp = 0;
        for (int u = 0; u < KNBR; ++u) {
          float v = s_bd[lane][u];
          if (v > mx) { mx = v; mp = u; }
        }
        worst = mx; wpos = mp;
      }
    }
    for (int t = 0; t < KNBR; ++t) {
      nbr[(size_t)qg * KNBR + t]   = s_bi[lane][t];
      d2out[(size_t)qg * KNBR + t] = s_bd[lane][t];
    }
  }
}

// ---------------------------------------------------------------------------
// Aggregate: for each node, weights = exp(-|scale*d2|); gather feat[nbr],
// out = [max_k, mean_k]  (width 2*NPROP). wave = node, lane = channel (P=32).
// ---------------------------------------------------------------------------
__global__ void aggregate_kernel(const float* __restrict__ feat,
                                 const int* __restrict__ nbr,
                                 const float* __restrict__ d2,
                                 const float* __restrict__ scale_ptr,
                                 float scale_const,
                                 float* __restrict__ out) {
  int lane = threadIdx.x & 31;
  int w    = threadIdx.x >> 5;
  int node = blockIdx.x * 8 + w;
  if (node >= NTOT) return;

  float scale = scale_ptr ? scale_ptr[0] : scale_const;
  const int*   nr = nbr + (size_t)node * KNBR;
  const float* dr = d2  + (size_t)node * KNBR;

  float vmax = -FLT_MAX, vsum = 0.f;
  for (int k = 0; k < KNBR; ++k) {
    int idx = nr[k];
    if (k + 1 < KNBR)  // hide gather latency: global_prefetch_b8 next row
      __builtin_prefetch(feat + (size_t)nr[k + 1] * NPROP, 0, 0);
    float wgt = __expf(-fabsf(scale * dr[k]));
    float v = feat[(size_t)idx * NPROP + lane] * wgt;
    vmax = fmaxf(vmax, v);
    vsum += v;
  }
  out[(size_t)node * (2 * NPROP) + lane]         = vmax;
  out[(size_t)node * (2 * NPROP) + NPROP + lane] = vsum * (1.f / KNBR);
}

// ---------------------------------------------------------------------------
extern "C" void kernel_launch(void* const* d_in, const int* in_sizes, int n_in,
                              void* d_out, int out_size, void* d_ws,
                              size_t ws_size, hipStream_t stream) {
  const float* x  = (const float*)d_in[0];
  // d_in[1]: row_splits (unused; segmentation is static)
  const float* Ws = (const float*)d_in[2];
  const float* bs = (const float*)d_in[3];
  const float* Wf = (const float*)d_in[4];
  const float* bf = (const float*)d_in[5];
  const float* Wm = (const float*)d_in[6];
  const float* bm = (const float*)d_in[7];
  const float* wd = (const float*)d_in[8];   // w_dist0 (1 element)
  const float* Wo = (const float*)d_in[9];
  const float* bo = (const float*)d_in[10];
  float* out = (float*)d_out;

  char* ws = (char*)d_ws;
  size_t off = 0;
  auto alloc = [&](size_t bytes) -> void* {
    void* p = ws + off;
    off = (off + bytes + 255) & ~(size_t)255;
    return p;
  };
  float* coords = (float*)alloc((size_t)NTOT * NDIM * sizeof(float));
  float* cnorm  = (float*)alloc((size_t)NTOT * sizeof(float));
  float* feat0  = (float*)alloc((size_t)NTOT * NPROP * sizeof(float));
  int*   nbrs   = (int*)  alloc((size_t)NTOT * KNBR * sizeof(int));
  float* d2     = (float*)alloc((size_t)NTOT * KNBR * sizeof(float));
  float* agg    = (float*)alloc((size_t)NTOT * 2 * NPROP * sizeof(float));
  float* feat1  = (float*)alloc((size_t)NTOT * NPROP * sizeof(float));
  float* agg2   = (float*)alloc((size_t)NTOT * 2 * NPROP * sizeof(float));

  // 1) coordinates = x @ W_s + b_s  (+ per-node squared norms)
  coords_kernel<<<(NTOT + 255) / 256, 256, 0, stream>>>(x, Ws, bs, coords,
                                                        cnorm);

  // 2) features = x @ W_flr + b_flr          (64 -> 32, linear)
  gemm_wmma_kernel<<<dim3(NTOT / 16, NPROP / 16), 32, 0, stream>>>(
      x, F_IN, nullptr, 0, Wf, bf, feat0, NPROP, /*act=*/0);

  // 3) per-event kNN (40 neighbours + their d2); one wave per 16 queries
  knn_kernel<<<NTOT / 16, 32, 0, stream>>>(coords, cnorm, nbrs, d2);

  // 4) aggregate pass 1: weights = exp(-10*d2)
  aggregate_kernel<<<NTOT / 8, 256, 0, stream>>>(feat0, nbrs, d2, nullptr,
                                                 10.0f, agg);

  // 5) features = elu([agg | x] @ W_mp0 + b)  (128 -> 32)
  gemm_wmma_kernel<<<dim3(NTOT / 16, NPROP / 16), 32, 0, stream>>>(
      agg, 2 * NPROP, x, F_IN, Wm, bm, feat1, NPROP, /*act=*/1);

  // 6) aggregate pass 2: weights = exp(-wd*d2)
  aggregate_kernel<<<NTOT / 8, 256, 0, stream>>>(feat1, nbrs, d2, wd, 1.0f,
                                                 agg2);

  // 7) out = tanh([x | agg2] @ W_out + b)     (128 -> 64)
  gemm_wmma_kernel<<<dim3(NTOT / 16, NFILT / 16), 32, 0, stream>>>(
      x, F_IN, agg2, 2 * NPROP, Wo, bo, out, NFILT, /*act=*/2);
}